// proj_to_tex_layer_61993557950628
// MI455X (gfx1250) — compile-verified
//
#include <hip/hip_runtime.h>

#define TEX_H 256
#define TEX_W 256
#define FCH   16
#define AF    2
#define KWIN  4
// DECAY = sqrt(2*0.5^2) = sqrt(0.5); 1/DECAY = sqrt(2)
#define INV_DECAY 1.41421356237309515f

typedef float v2f __attribute__((ext_vector_type(2)));
typedef float v8f __attribute__((ext_vector_type(8)));

__device__ __forceinline__ void atomic_add_f32(float* p, float v) {
  // relaxed, device scope -> native global_atomic_add_f32 on gfx1250 (L2 atomic)
  __hip_atomic_fetch_add(p, v, __ATOMIC_RELAXED, __HIP_MEMORY_SCOPE_AGENT);
}

// ---------------- kernel 1: zero workspace (wsum + csum) ----------------
__global__ __launch_bounds__(256) void zero_ws_kernel(float4* __restrict__ ws, int n4) {
  int i = blockIdx.x * blockDim.x + threadIdx.x;
  int stride = gridDim.x * blockDim.x;
  float4 z = make_float4(0.f, 0.f, 0.f, 0.f);
  for (; i < n4; i += stride) ws[i] = z;
}

// ---------------- kernel 2: gated atomic splat ----------------
// One thread per source pixel. Weights with w <= 0.1 (or mask==0, or
// out-of-range) contribute exactly zero in the reference, so we skip
// their atomics entirely (~4x fewer L2 atomics).
__global__ __launch_bounds__(256) void splat_kernel(
    const float* __restrict__ f_map, const float* __restrict__ uv_map,
    const float* __restrict__ mask, float* __restrict__ wsum,
    float* __restrict__ csum, int N) {
  int p = blockIdx.x * blockDim.x + threadIdx.x;
  if (p >= N) return;
  float m = mask[p];
  if (m == 0.0f) return;

  float2 uv = ((const float2*)uv_map)[p];
  float u = uv.x * (float)TEX_W;
  float v = uv.y * (float)TEX_H;
  float us = fmaxf(u - (float)AF, 0.0f);
  float vs = fmaxf(v - (float)AF, 0.0f);

  const float4* pix4 = ((const float4*)f_map) + (size_t)p * 4;
  float4 c0 = pix4[0], c1 = pix4[1], c2 = pix4[2], c3 = pix4[3];

#pragma unroll
  for (int j = 0; j < KWIN; ++j) {
    float vg = vs + (float)j;
    float dv = fabsf(vg - v);
    dv = fminf(dv, fabsf(dv - (float)TEX_H));   // wrap distance (matches ref)
    int vi = (int)vg;
#pragma unroll
    for (int i = 0; i < KWIN; ++i) {
      float ug = us + (float)i;
      float du = fabsf(ug - u);
      du = fminf(du, fabsf(du - (float)TEX_W));
      float w = __expf(-sqrtf(du * du + dv * dv) * INV_DECAY);
      int ui = (int)ug;
      if (w > 0.1f && vi < TEX_H && ui < TEX_W) {
        int idx = vi * TEX_W + ui;
        atomic_add_f32(&wsum[idx], w);
        float* cs = &csum[(size_t)idx * FCH];
        atomic_add_f32(cs + 0,  w * c0.x); atomic_add_f32(cs + 1,  w * c0.y);
        atomic_add_f32(cs + 2,  w * c0.z); atomic_add_f32(cs + 3,  w * c0.w);
        atomic_add_f32(cs + 4,  w * c1.x); atomic_add_f32(cs + 5,  w * c1.y);
        atomic_add_f32(cs + 6,  w * c1.z); atomic_add_f32(cs + 7,  w * c1.w);
        atomic_add_f32(cs + 8,  w * c2.x); atomic_add_f32(cs + 9,  w * c2.y);
        atomic_add_f32(cs + 10, w * c2.z); atomic_add_f32(cs + 11, w * c2.w);
        atomic_add_f32(cs + 12, w * c3.x); atomic_add_f32(cs + 13, w * c3.y);
        atomic_add_f32(cs + 14, w * c3.z); atomic_add_f32(cs + 15, w * c3.w);
      }
    }
  }
}

// ---------------- kernel 3: WMMA normalize + broadcast ----------------
// Per wave32: 4 texels. D(16x16) = A(16x4) x B(4x16) with
//   B[k][n] = csum[t0+k][n]           (k = texel-in-group, n = channel)
//   A[m][k] = s_k * delta(m&3, k)     (s_k = (wsum>0.01)/(wsum+0.001))
// => D[m][n] = s * csum[t0+(m&3)][n], each texel row replicated 4x across D;
// the 4 replicas each store 2 of the 8 broadcast batches.
// A layout (ISA 7.12.2, 32-bit A 16x4): lanes 0-15 hold K=0(V0),K=1(V1);
// lanes 16-31 hold K=2(V0),K=3(V1). B mirrors: lanes 0-15 rows 0/1,
// lanes 16-31 rows 2/3. C/D: lane holds col n=lane&15, V_j = row j+8*(lane>>4).
__global__ __launch_bounds__(256) void normalize_wmma_kernel(
    const float* __restrict__ wsum, const float* __restrict__ csum,
    float* __restrict__ out) {
  int lane = threadIdx.x & 31;
  int wave = threadIdx.x >> 5;
  int t0   = (blockIdx.x * 8 + wave) * 4;   // 4 texels per wave
  int half = lane >> 4;
  int n    = lane & 15;                     // channel (and row index m for A)
  int m    = lane & 15;

  v2f b;
  b.x = csum[(size_t)(t0 + 2 * half + 0) * FCH + n];
  b.y = csum[(size_t)(t0 + 2 * half + 1) * FCH + n];

  int   k0 = 2 * half, k1 = 2 * half + 1;
  float w0 = wsum[t0 + k0];
  float w1 = wsum[t0 + k1];
  float s0 = (w0 > 0.01f ? 1.0f : 0.0f) / (w0 + 0.001f);
  float s1 = (w1 > 0.01f ? 1.0f : 0.0f) / (w1 + 0.001f);

  v2f a;
  a.x = ((m & 3) == k0) ? s0 : 0.0f;
  a.y = ((m & 3) == k1) ? s1 : 0.0f;

  v8f c = {0.f, 0.f, 0.f, 0.f, 0.f, 0.f, 0.f, 0.f};
  v8f d = __builtin_amdgcn_wmma_f32_16x16x4_f32(
      /*neg_a=*/false, a, /*neg_b=*/false, b,
      /*c_mod=*/(short)0, c, /*reuse_a=*/false, /*reuse_b=*/false);

  // row m' = j + 8*half -> texel t0 + (j&3); replica r = (j>>2) + 2*half
#pragma unroll
  for (int j = 0; j < 8; ++j) {
    int   t   = t0 + (j & 3);
    int   r   = (j >> 2) + 2 * half;
    float val = d[j];
    out[((size_t)(2 * r + 0) * (TEX_H * TEX_W) + t) * FCH + n] = val;
    out[((size_t)(2 * r + 1) * (TEX_H * TEX_W) + t) * FCH + n] = val;
  }
}

extern "C" void kernel_launch(void* const* d_in, const int* in_sizes, int n_in,
                              void* d_out, int out_size, void* d_ws, size_t ws_size,
                              hipStream_t stream) {
  const float* f_map  = (const float*)d_in[0];
  const float* uv_map = (const float*)d_in[1];
  const float* mask   = (const float*)d_in[2];
  float* out = (float*)d_out;

  // workspace: wsum [65536] then csum [65536*16]  (4.45 MB total)
  float* wsum = (float*)d_ws;
  float* csum = wsum + TEX_H * TEX_W;

  int n4 = (TEX_H * TEX_W * (FCH + 1)) / 4;
  zero_ws_kernel<<<512, 256, 0, stream>>>((float4*)d_ws, n4);

  int N = in_sizes[2];  // B*H*W = 524288 (mask element count)
  splat_kernel<<<(N + 255) / 256, 256, 0, stream>>>(f_map, uv_map, mask, wsum, csum, N);

  // 4 texels/wave * 8 waves/block = 32 texels/block; 65536/32 = 2048 blocks
  normalize_wmma_kernel<<<(TEX_H * TEX_W) / 32, 256, 0, stream>>>(wsum, csum, out);
}